// SaliencyLoss_10419590660539
// MI455X (gfx1250) — compile-verified
//
#include <hip/hip_runtime.h>
#include <cstdint>
#include <cstddef>

#define IMG_H 512
#define IMG_W 512
#define N_IMG 16
#define HW    (IMG_H * IMG_W)
#define TILE  32
#define TPB   256
#define HDIM  34
#define HN    (HDIM * HDIM)          // 1156 haloed gray elements
#define CHB   (HN * 4)               // bytes per channel tile in LDS
#define EPSF  1e-8f

// ---------------------------------------------------------------------------
// CDNA5 async global->LDS copy (VGLOBAL GLOBAL_LOAD_ASYNC_TO_LDS_B32, GV mode)
// Per-lane: copies 4 bytes from global address g to LDS offset lds_off; tracked by ASYNCcnt.
// ---------------------------------------------------------------------------
static __device__ __forceinline__ void async_b32(uint32_t lds_off, const float* g) {
    asm volatile("global_load_async_to_lds_b32 %0, %1, off"
                 :: "v"(lds_off), "v"((uint64_t)(uintptr_t)g)
                 : "memory");
}
static __device__ __forceinline__ void wait_async0() {
    asm volatile("s_wait_asynccnt 0" ::: "memory");
}

static __device__ __forceinline__ int reflect101(int x, int n) {
    return x < 0 ? -x : (x >= n ? 2 * n - 2 - x : x);
}

// Exact-sequence grayscale (no fma contraction) then uint8 truncation, like
// XLA: q = float(uint8((w0*r + w1*g + w2*b) * 255))
static __device__ __forceinline__ float quant_gray(float r, float g, float b) {
    float s = __fadd_rn(__fadd_rn(__fmul_rn(0.2989f, r), __fmul_rn(0.587f, g)),
                        __fmul_rn(0.114f, b));
    return truncf(__fmul_rn(s, 255.0f));
}

// 3x3 Sobel magnitude; g is 34x34 haloed quantized-gray tile, (ly,lx) in [0,32).
// All values are small integers -> exact in fp32 regardless of op ordering.
static __device__ __forceinline__ float sobel_mag(const float* g, int ly, int lx) {
    const float* r0 = g + ly * HDIM + lx;
    const float* r1 = r0 + HDIM;
    const float* r2 = r1 + HDIM;
    float a00 = r0[0], a01 = r0[1], a02 = r0[2];
    float a10 = r1[0],              a12 = r1[2];
    float a20 = r2[0], a21 = r2[1], a22 = r2[2];
    float gx = (a02 + 2.0f * a12 + a22) - (a00 + 2.0f * a10 + a20);
    float gy = (a20 + 2.0f * a21 + a22) - (a00 + 2.0f * a01 + a02);
    return sqrtf(gx * gx + gy * gy);
}

// ---------------------------------------------------------------------------
// k_init: reset workspace (harness poisons ws once; must re-init every call)
// ---------------------------------------------------------------------------
__global__ void k_init(uint32_t* mm, double* acc) {
    int t = threadIdx.x;
    if (t < 64) mm[t] = (t & 1) ? 0u : 0xFFFFFFFFu;   // [min,max] pairs (u32 bits)
    if (t == 0) *acc = 0.0;
}

// ---------------------------------------------------------------------------
// k_minmax: per-image min/max of Sobel magnitude for vis (set 0) and ir (set 1)
// grid = 2 sets * 16 images * 256 tiles = 8192 blocks of 256 threads
// ---------------------------------------------------------------------------
__global__ void __launch_bounds__(TPB) k_minmax(const float* __restrict__ vis,
                                                const float* __restrict__ ir,
                                                uint32_t* __restrict__ mm) {
    __shared__ float ch[3][HN];
    __shared__ float gray[HN];
    __shared__ float redmn[8];
    __shared__ float redmx[8];

    int bid  = blockIdx.x;
    int set  = bid >> 12;
    int rem  = bid & 4095;
    int img  = rem >> 8;
    int tile = rem & 255;
    int i0 = (tile >> 4) * TILE;
    int j0 = (tile & 15) * TILE;

    const float* src = (set == 0 ? vis : ir) + (size_t)img * 3 * HW;
    uint32_t lds0 = (uint32_t)(size_t)&ch[0][0];   // low 32 bits of flat addr = LDS offset

    // stage haloed RGB tile into LDS via async copies (reflect-101 addressing)
    for (int s = threadIdx.x; s < HN; s += TPB) {
        int ty = s / HDIM, tx = s - ty * HDIM;
        int gi = reflect101(i0 - 1 + ty, IMG_H);
        int gj = reflect101(j0 - 1 + tx, IMG_W);
        const float* p = src + (size_t)gi * IMG_W + gj;
        uint32_t lo = lds0 + (uint32_t)s * 4u;
        async_b32(lo,            p);
        async_b32(lo + CHB,      p + HW);
        async_b32(lo + 2u * CHB, p + 2 * HW);
    }
    wait_async0();
    __syncthreads();

    for (int s = threadIdx.x; s < HN; s += TPB)
        gray[s] = quant_gray(ch[0][s], ch[1][s], ch[2][s]);
    __syncthreads();

    float mn = 3.0e38f, mx = 0.0f;
    int lx  = threadIdx.x & 31;
    int ly0 = threadIdx.x >> 5;       // 0..7
    for (int k = 0; k < 4; ++k) {
        float m = sobel_mag(gray, ly0 + 8 * k, lx);
        mn = fminf(mn, m);
        mx = fmaxf(mx, m);
    }
    // wave32 reduction
    for (int off = 16; off > 0; off >>= 1) {
        mn = fminf(mn, __shfl_xor(mn, off, 32));
        mx = fmaxf(mx, __shfl_xor(mx, off, 32));
    }
    int wave = threadIdx.x >> 5;
    if ((threadIdx.x & 31) == 0) { redmn[wave] = mn; redmx[wave] = mx; }
    __syncthreads();
    if (threadIdx.x == 0) {
        for (int w = 1; w < 8; ++w) {
            mn = fminf(mn, redmn[w]);
            mx = fmaxf(mx, redmx[w]);
        }
        // mag >= 0, so u32 bit-pattern order == float order
        uint32_t* slot = mm + ((set << 4) + img) * 2;
        atomicMin(slot,     __float_as_uint(mn));
        atomicMax(slot + 1, __float_as_uint(mx));
    }
}

// ---------------------------------------------------------------------------
// k_loss: normalize, blend, accumulate mean-L1 numerator
// grid = 16 images * 256 tiles = 4096 blocks of 256 threads
// ---------------------------------------------------------------------------
__global__ void __launch_bounds__(TPB) k_loss(const float* __restrict__ vis,
                                              const float* __restrict__ ir,
                                              const float* __restrict__ fus,
                                              const uint32_t* __restrict__ mm,
                                              double* __restrict__ acc) {
    __shared__ float chV[3][HN];
    __shared__ float chI[3][HN];
    __shared__ float gV[HN];
    __shared__ float gI[HN];
    __shared__ double dred[8];

    int bid  = blockIdx.x;
    int img  = bid >> 8;
    int tile = bid & 255;
    int i0 = (tile >> 4) * TILE;
    int j0 = (tile & 15) * TILE;

    size_t base = (size_t)img * 3 * HW;
    uint32_t ldsV = (uint32_t)(size_t)&chV[0][0];
    uint32_t ldsI = (uint32_t)(size_t)&chI[0][0];

    for (int s = threadIdx.x; s < HN; s += TPB) {
        int ty = s / HDIM, tx = s - ty * HDIM;
        int gi = reflect101(i0 - 1 + ty, IMG_H);
        int gj = reflect101(j0 - 1 + tx, IMG_W);
        size_t off = base + (size_t)gi * IMG_W + gj;
        const float* pv = vis + off;
        const float* pi = ir + off;
        uint32_t lo = (uint32_t)s * 4u;
        async_b32(ldsV + lo,            pv);
        async_b32(ldsV + CHB + lo,      pv + HW);
        async_b32(ldsV + 2u * CHB + lo, pv + 2 * HW);
        async_b32(ldsI + lo,            pi);
        async_b32(ldsI + CHB + lo,      pi + HW);
        async_b32(ldsI + 2u * CHB + lo, pi + 2 * HW);
    }
    wait_async0();
    __syncthreads();

    for (int s = threadIdx.x; s < HN; s += TPB) {
        gV[s] = quant_gray(chV[0][s], chV[1][s], chV[2][s]);
        gI[s] = quant_gray(chI[0][s], chI[1][s], chI[2][s]);
    }
    __syncthreads();

    float mnV = __uint_as_float(mm[img * 2 + 0]);
    float mxV = __uint_as_float(mm[img * 2 + 1]);
    float mnI = __uint_as_float(mm[(16 + img) * 2 + 0]);
    float mxI = __uint_as_float(mm[(16 + img) * 2 + 1]);
    float dV = fmaxf(mxV - mnV, EPSF);
    float dI = fmaxf(mxI - mnI, EPSF);

    int lx  = threadIdx.x & 31;
    int ly0 = threadIdx.x >> 5;
    double lacc = 0.0;
    for (int k = 0; k < 4; ++k) {
        int ly = ly0 + 8 * k;
        // true IEEE divisions to match the reference exactly
        float sv = (sobel_mag(gV, ly, lx) - mnV) / dV;
        float si = (sobel_mag(gI, ly, lx) - mnI) / dI;
        float den = sv + si + EPSF;
        float wv = sv / den;
        float wi = si / den;
        int sl = (ly + 1) * HDIM + (lx + 1);         // center pixel in haloed tile
        size_t gidx = base + (size_t)(i0 + ly) * IMG_W + (j0 + lx);
        float a = 0.0f;
        #pragma unroll
        for (int c = 0; c < 3; ++c) {
            float pv = chV[c][sl];                    // RGB already staged in LDS
            float pi = chI[c][sl];
            float pf = __builtin_nontemporal_load(fus + gidx + (size_t)c * HW);
            float sal = __fadd_rn(__fmul_rn(wv, pv), __fmul_rn(wi, pi));
            a += fabsf(__fadd_rn(sal, -pf));
        }
        lacc += (double)a;
    }
    for (int off = 16; off > 0; off >>= 1)
        lacc += __shfl_xor(lacc, off, 32);
    int wave = threadIdx.x >> 5;
    if ((threadIdx.x & 31) == 0) dred[wave] = lacc;
    __syncthreads();
    if (threadIdx.x == 0) {
        double t = dred[0];
        for (int w = 1; w < 8; ++w) t += dred[w];
        atomicAdd(acc, t);                            // global_atomic_add_f64
    }
}

// ---------------------------------------------------------------------------
__global__ void k_final(const double* acc, float* out) {
    out[0] = (float)(acc[0] / (double)((size_t)N_IMG * 3 * HW));
}

extern "C" void kernel_launch(void* const* d_in, const int* in_sizes, int n_in,
                              void* d_out, int out_size, void* d_ws, size_t ws_size,
                              hipStream_t stream) {
    const float* vis = (const float*)d_in[0];
    const float* ir  = (const float*)d_in[1];
    const float* fus = (const float*)d_in[2];
    float* out = (float*)d_out;
    uint32_t* mm = (uint32_t*)d_ws;                       // 64 u32: [2 sets][16 imgs][min,max]
    double* acc  = (double*)((char*)d_ws + 256);          // f64 accumulator

    hipLaunchKernelGGL(k_init, dim3(1), dim3(64), 0, stream, mm, acc);
    hipLaunchKernelGGL(k_minmax, dim3(2 * N_IMG * 256), dim3(TPB), 0, stream, vis, ir, mm);
    hipLaunchKernelGGL(k_loss, dim3(N_IMG * 256), dim3(TPB), 0, stream, vis, ir, fus, mm, acc);
    hipLaunchKernelGGL(k_final, dim3(1), dim3(1), 0, stream, acc, out);
}